// EmbedLayer_14611478741481
// MI455X (gfx1250) — compile-verified
//
#include <hip/hip_runtime.h>
#include <cstdint>
#include <cstddef>

// Embedding fusion for MI455X (gfx1250, wave32):
//   out[b,s,:] = token_table[tok] + segm_table[seg] + sinusoidal_pos(s)
// Memory-bound (~100 MB HBM traffic -> ~4.3us floor @ 23.3 TB/s).
// CDNA5 path used: async global->LDS copy (ASYNCcnt) to overlap the random
// token-row gather with the transcendental positional-encoding math.

#define VOCAB 32000
#define DIM   768
#define BATCH 8
#define SEQ   2048
#define ROWS  (BATCH * SEQ)
#define THREADS_PER_ROW (DIM / 4)   /* 192 lanes = 6 wave32 waves, 1 float4 each */

#define LOG2_10000 13.287712379549449f
#define INV_2PI    0.15915494309189535f
#define INV_DIM    (1.0f / (float)DIM)

__global__ __launch_bounds__(THREADS_PER_ROW)
void embed_fused_kernel(const int* __restrict__ tokens,        // (B*S)
                        const int* __restrict__ segments,      // (B*S)
                        const float* __restrict__ token_table, // (VOCAB, DIM)
                        const float* __restrict__ segm_table,  // (3, DIM)
                        float* __restrict__ out)               // (B*S, DIM)
{
    __shared__ __align__(16) float lds_tok[DIM];

    const int row = blockIdx.x;            // b*SEQ + s
    const int s   = row & (SEQ - 1);       // SEQ is a power of two
    const int t   = threadIdx.x;           // 0..191
    const int id0 = t << 2;                // first of this lane's 4 dim ids

    const int tok = tokens[row];           // uniform across block
    const int seg = segments[row];         // uniform across block

    // ---- 1) kick off async token-row gather: HBM/L2 -> LDS, 16 B/lane ----
    // VGLOBAL async encoding: VDST = LDS byte-address VGPR, VADDR = 64-bit
    // global address pair, no SADDR ("off"). Tracked by ASYNCcnt.
    {
        const float* gsrc = token_table + (size_t)(unsigned)tok * DIM + id0;
        uint32_t lds_off = (uint32_t)(uintptr_t)&lds_tok[id0];  // low 32 bits of
                                                                // generic = LDS offset
        uint64_t gaddr   = (uint64_t)(uintptr_t)gsrc;
        asm volatile("global_load_async_to_lds_b128 %0, %1, off"
                     :: "v"(lds_off), "v"(gaddr) : "memory");
    }

    // ---- 2) overlap: sinusoidal positional encoding (VALU/trans only) ----
    // even id 2k:   sin(pos * 10000^(-2k/D));  odd id 2k+1: cos(same angle)
    // hw v_sin/v_cos take turns (sin(2*pi*x)) -> scale by 1/2pi, fract-reduce.
    const float pf = (float)s;
    const float e0 = (float)id0       * INV_DIM;
    const float e1 = (float)(id0 + 2) * INV_DIM;
    const float f0 = __builtin_amdgcn_exp2f(-e0 * LOG2_10000);  // 10000^-e0
    const float f1 = __builtin_amdgcn_exp2f(-e1 * LOG2_10000);  // 10000^-e1
    const float t0 = __builtin_amdgcn_fractf(pf * f0 * INV_2PI);
    const float t1 = __builtin_amdgcn_fractf(pf * f1 * INV_2PI);
    const float p0 = __builtin_amdgcn_sinf(t0);   // sin(angle0)
    const float p1 = __builtin_amdgcn_cosf(t0);   // cos(angle0)
    const float p2 = __builtin_amdgcn_sinf(t1);   // sin(angle1)
    const float p3 = __builtin_amdgcn_cosf(t1);   // cos(angle1)

    // ---- 3) segment row (only 3 rows exist -> always cache-hot) ----
    // Written before the wait + memory clobber pins it before the stall point.
    const float4 sg = *(const float4*)(segm_table + seg * DIM + id0);

    // ---- 4) drain async copy, read token row back from LDS ----
    // The dummy "v" inputs create data deps on the trans results so the wait
    // cannot be scheduled before the positional math -> real overlap window.
    asm volatile("s_wait_asynccnt 0"
                 :: "v"(p0), "v"(p1), "v"(p2), "v"(p3)
                 : "memory");
    const float4 tk = *(const float4*)&lds_tok[id0];  // lane reads its own slot

    float4 o;
    o.x = sg.x + p0 + tk.x;
    o.y = sg.y + p1 + tk.y;
    o.z = sg.z + p2 + tk.z;
    o.w = sg.w + p3 + tk.w;
    *(float4*)(out + (size_t)row * DIM + id0) = o;    // global_store_b128
}

extern "C" void kernel_launch(void* const* d_in, const int* in_sizes, int n_in,
                              void* d_out, int out_size, void* d_ws, size_t ws_size,
                              hipStream_t stream) {
    (void)in_sizes; (void)n_in; (void)out_size; (void)d_ws; (void)ws_size;
    const int*   tokens      = (const int*)  d_in[0];  // input_tensor (B,S) int32
    const int*   segments    = (const int*)  d_in[1];  // segment_label (B,S) int32
    const float* token_table = (const float*)d_in[2];  // (VOCAB, DIM) f32
    const float* segm_table  = (const float*)d_in[3];  // (3, DIM) f32
    float*       out         = (float*)d_out;          // (B,S,DIM) f32

    dim3 grid(ROWS);
    dim3 block(THREADS_PER_ROW);
    hipLaunchKernelGGL(embed_fused_kernel, grid, block, 0, stream,
                       tokens, segments, token_table, segm_table, out);
}